// SSD_loss_89627377532963
// MI455X (gfx1250) — compile-verified
//
#include <hip/hip_runtime.h>
#include <hip/hip_bf16.h>

#define BNUM 64
#define PNUM 8732
#define MNUM 16
#define CNUM 21

typedef __attribute__((ext_vector_type(2))) float v2f;
typedef __attribute__((ext_vector_type(8))) float v8f;

__device__ __constant__ float CLASS_W[CNUM] = {
    0.5f, 1.0f, 1.0f, 2.5f, 2.5f, 2.5f, 1.0f, 0.75f, 0.75f, 2.5f, 2.5f,
    2.5f, 0.75f, 1.0f, 1.0f, 0.75f, 2.5f, 2.5f, 2.5f, 1.0f, 1.0f};

__device__ __forceinline__ float smooth_l1(float d) {
  float a = fabsf(d);
  return (a < 1.0f) ? 0.5f * a * a : a - 0.5f;
}

// ---------------- Kernel A: per-row matching -------------------------------
// Computes, per prior: best truth (argmax over 16, first-max tie-break) and
// whether IoU >= 0.5 -> packed int8 code m in {-1, 0..15}.
// Also per truth: argmax prior over P (smallest-index tie-break).
__global__ __launch_bounds__(256) void match_kernel(
    const float* __restrict__ defaults, const float* __restrict__ gt_boxes,
    signed char* __restrict__ m_out, int* __restrict__ bp_out) {
  const int b = blockIdx.x, tid = threadIdx.x;
  __shared__ float tb[MNUM][4];
  __shared__ float area2[MNUM];
  __shared__ float redv[256][MNUM];
  __shared__ int redi[256][MNUM];

  if (tid < MNUM * 4) ((float*)tb)[tid] = gt_boxes[b * MNUM * 4 + tid];
  __syncthreads();
  if (tid < MNUM)
    area2[tid] = (tb[tid][2] - tb[tid][0]) * (tb[tid][3] - tb[tid][1]);
  __syncthreads();

  float bov[MNUM];
  int bix[MNUM];
#pragma unroll
  for (int j = 0; j < MNUM; ++j) { bov[j] = -1.0f; bix[j] = 0; }

  for (int p = tid; p < PNUM; p += 256) {
    const float d0 = defaults[p * 4 + 0], d1 = defaults[p * 4 + 1];
    const float d2 = defaults[p * 4 + 2], d3 = defaults[p * 4 + 3];
    const float a1 = (d2 - d0) * (d3 - d1);
    float bo = -1.0f;
    int bi = 0;
#pragma unroll
    for (int j = 0; j < MNUM; ++j) {
      float lx = fmaxf(d0, tb[j][0]), ly = fmaxf(d1, tb[j][1]);
      float rx = fminf(d2, tb[j][2]), ry = fminf(d3, tb[j][3]);
      float w = fmaxf(rx - lx, 0.0f), h = fmaxf(ry - ly, 0.0f);
      float inter = w * h;
      float iou = inter / (a1 + area2[j] - inter);
      if (iou > bo) { bo = iou; bi = j; }            // first max over j
      if (iou > bov[j]) { bov[j] = iou; bix[j] = p; } // first max over p (per thread)
    }
    m_out[(size_t)b * PNUM + p] = (bo < 0.5f) ? (signed char)-1 : (signed char)bi;
  }

#pragma unroll
  for (int j = 0; j < MNUM; ++j) { redv[tid][j] = bov[j]; redi[tid][j] = bix[j]; }
  __syncthreads();
  for (int off = 128; off >= 1; off >>= 1) {
    for (int k = tid; k < off * MNUM; k += 256) {
      int t = k >> 4, j = k & 15;
      float v1 = redv[t][j], v2 = redv[t + off][j];
      int i1 = redi[t][j], i2 = redi[t + off][j];
      if (v2 > v1 || (v2 == v1 && i2 < i1)) { redv[t][j] = v2; redi[t][j] = i2; }
    }
    __syncthreads();
  }
  if (tid < MNUM) bp_out[b * MNUM + tid] = redi[0][tid];
}

// ---------------- Kernel A2: sequential override ---------------------------
__global__ void override_kernel(signed char* __restrict__ m,
                                const int* __restrict__ bp) {
  const int b = blockIdx.x;
  for (int j = 0; j < MNUM; ++j)  // ascending j: later truths win, like the ref
    m[(size_t)b * PNUM + bp[b * MNUM + j]] = (signed char)j;
}

// ---------------- Kernel BC: fused loss partials + hard-neg top-k ----------
__global__ __launch_bounds__(1024) void loss_kernel(
    const float* __restrict__ loc_preds, const float* __restrict__ conf_preds,
    const float* __restrict__ cent_preds, const float* __restrict__ defaults,
    const float* __restrict__ gt_boxes, const int* __restrict__ gt_labels,
    const signed char* __restrict__ m_in, float* __restrict__ rows) {
  const int b = blockIdx.x, tid = threadIdx.x;
  __shared__ float lossc[PNUM];
  __shared__ float tb[MNUM][4];
  __shared__ int tl[MNUM];
  __shared__ float wsum[32][4][16];
  __shared__ float rowpart[4];
  __shared__ unsigned s_cnt;
  __shared__ float s_fsum;

  if (tid < MNUM * 4) ((float*)tb)[tid] = gt_boxes[b * MNUM * 4 + tid];
  if (tid >= 64 && tid < 64 + MNUM) tl[tid - 64] = gt_labels[b * MNUM + tid - 64];
  __syncthreads();

  float lacc = 0.0f, bacc = 0.0f, facc = 0.0f, pcnt = 0.0f;
  const float* cb = conf_preds + (size_t)b * PNUM * CNUM;

  for (int p = tid; p < PNUM; p += 1024) {
    if (p + 1024 < PNUM)
      __builtin_prefetch(cb + (size_t)(p + 1024) * CNUM, 0, 0);  // global_prefetch_b8
    const int mm = (int)m_in[(size_t)b * PNUM + p];
    const float* x = cb + (size_t)p * CNUM;
    float xv[CNUM];
#pragma unroll
    for (int c = 0; c < CNUM; ++c) xv[c] = x[c];
    float mx = xv[0];
#pragma unroll
    for (int c = 1; c < CNUM; ++c) mx = fmaxf(mx, xv[c]);
    float se = 0.0f;
#pragma unroll
    for (int c = 0; c < CNUM; ++c) se += expf(xv[c] - mx);
    const float lse = mx + logf(se);
    const int t = (mm < 0) ? 0 : tl[mm];
    const float ce = lse - xv[t];
    const float pt = expf(-ce);
    const float omp = 1.0f - pt;
    const float focal = 0.25f * omp * omp * CLASS_W[t] * ce;
    const bool pos = (mm >= 0);
    lossc[p] = pos ? 0.0f : focal;
    if (pos) {
      facc += focal;
      pcnt += 1.0f;
      const float g0 = tb[mm][0], g1 = tb[mm][1], g2 = tb[mm][2], g3 = tb[mm][3];
      const float d0 = defaults[p * 4 + 0], d1 = defaults[p * 4 + 1];
      const float d2 = defaults[p * 4 + 2], d3 = defaults[p * 4 + 3];
      const float gw = g2 - g0, gh = g3 - g1;
      const float gcx = g0 + gw * 0.5f, gcy = g1 + gh * 0.5f;
      const float dw = d2 - d0, dh = d3 - d1;
      const float dcx = d0 + dw * 0.5f, dcy = d1 + dh * 0.5f;
      const float e0 = (gcx - dcx) / (dw * 0.1f + 1e-8f);
      const float e1 = (gcy - dcy) / (dh * 0.1f + 1e-8f);
      const float e2 = logf(gw / (dw + 1e-8f) + 1e-8f) / 0.1f;
      const float e3 = logf(gh / (dh + 1e-8f) + 1e-8f) / 0.1f;
      const float* lp = loc_preds + ((size_t)b * PNUM + p) * 4;
      lacc += smooth_l1(lp[0] - e0) + smooth_l1(lp[1] - e1) +
              smooth_l1(lp[2] - e2) + smooth_l1(lp[3] - e3);
      const float cx = (g0 + g2) * 0.5f, cy = (g1 + g3) * 0.5f;
      const float ld = cx - g0, rd = g2 - cx, td = cy - g1, bd = g3 - cy;
      const float rxr = fminf(ld, rd) / fmaxf(ld, rd + 1e-8f);
      const float ryr = fminf(td, bd) / fmaxf(td, bd + 1e-8f);
      const float ct = sqrtf(rxr * ryr);
      const float xc = cent_preds[(size_t)b * PNUM + p];
      bacc += fmaxf(xc, 0.0f) - xc * ct + log1pf(expf(-fabsf(xc)));
    }
  }

  // ---- WMMA wave reduction: D = ones(16x4) x B -> column sums; the total of
  // all 64 B entries is layout-agnostic (sum of all column sums). One WMMA per
  // accumulator per wave; lane's value in B vgpr0, vgpr1 = 0.
  {
    const int wv = tid >> 5, lane = tid & 31;
    v2f ones; ones.x = 1.0f; ones.y = 1.0f;
    float accs4[4] = {lacc, bacc, facc, pcnt};
#pragma unroll
    for (int a = 0; a < 4; ++a) {
      v2f bm; bm.x = accs4[a]; bm.y = 0.0f;
      v8f cz = {};
      v8f dr = __builtin_amdgcn_wmma_f32_16x16x4_f32(
          false, ones, false, bm, (short)0, cz, false, false);
      if (lane < 16) wsum[wv][a][lane] = dr[0];  // colsums replicated per row
    }
  }
  __syncthreads();
  if (tid < 4) {
    float s = 0.0f;
    for (int w = 0; w < 32; ++w)
      for (int l = 0; l < 16; ++l) s += wsum[w][tid][l];
    rowpart[tid] = s;
  }
  __syncthreads();

  const float posr = rowpart[3];
  int k = 3 * (int)(posr + 0.5f);
  if (k > PNUM - 1) k = PNUM - 1;

  // ---- exact k-th-largest selection over lossc (all values >= 0) ----
  auto countG = [&](float thr) -> unsigned {
    if (tid == 0) s_cnt = 0u;
    __syncthreads();
    unsigned c = 0;
    for (int p = tid; p < PNUM; p += 1024) c += (lossc[p] > thr) ? 1u : 0u;
    atomicAdd(&s_cnt, c);
    __syncthreads();
    unsigned r = s_cnt;
    __syncthreads();
    return r;
  };

  float negsum = 0.0f;
  if (k > 0) {
    const unsigned c0 = countG(0.0f);
    float v;
    if ((int)c0 < k) {
      v = 0.0f;  // fewer than k strictly-positive values: take them all
    } else {
      unsigned lo = 0u, hi = 0x7F800000u;  // c(lo)>=k, c(hi)<k
      while (hi - lo > 1u) {
        unsigned mid = (lo + hi) >> 1;
        unsigned c = countG(__uint_as_float(mid));
        if ((int)c >= k) lo = mid; else hi = mid;
      }
      v = __uint_as_float(hi);  // exact k-th largest value
    }
    if (tid == 0) { s_cnt = 0u; s_fsum = 0.0f; }
    __syncthreads();
    unsigned cg = 0; float sg = 0.0f;
    for (int p = tid; p < PNUM; p += 1024) {
      float xval = lossc[p];
      if (xval > v) { cg++; sg += xval; }
    }
    atomicAdd(&s_cnt, cg);
    atomicAdd(&s_fsum, sg);
    __syncthreads();
    int total_g = (int)s_cnt;
    float sum_g = s_fsum;
    negsum = (total_g < k) ? (sum_g + (float)(k - total_g) * v) : sum_g;
  }

  if (tid == 0) {
    float* r = rows + b * 6;
    r[0] = rowpart[0];      // smooth-L1 sum (pos)
    r[1] = rowpart[1];      // BCE sum (pos)
    r[2] = rowpart[2];      // focal sum (pos)
    r[3] = rowpart[3];      // pos count
    r[4] = negsum;          // top-k negative focal sum
    r[5] = (float)k;        // neg count
  }
}

// ---------------- Kernel D: final combine ----------------------------------
__global__ __launch_bounds__(64) void finalize_kernel(
    const float* __restrict__ rows, float* __restrict__ out) {
  __shared__ float acc[6][64];
  const int tid = threadIdx.x;
  for (int i = 0; i < 6; ++i) acc[i][tid] = rows[tid * 6 + i];
  __syncthreads();
  for (int off = 32; off >= 1; off >>= 1) {
    if (tid < off)
      for (int i = 0; i < 6; ++i) acc[i][tid] += acc[i][tid + off];
    __syncthreads();
  }
  if (tid == 0) {
    const float npos = acc[3][0];
    const float loc = acc[0][0] / npos;
    const float cent = acc[1][0] / npos;
    const float conf = (acc[2][0] + acc[4][0]) / (npos + acc[5][0]);
    out[0] = 2.0f * loc + conf + cent;
  }
}

extern "C" void kernel_launch(void* const* d_in, const int* in_sizes, int n_in,
                              void* d_out, int out_size, void* d_ws,
                              size_t ws_size, hipStream_t stream) {
  const float* loc_preds = (const float*)d_in[0];     // (64,8732,4)
  const float* conf_preds = (const float*)d_in[1];    // (64,8732,21)
  const float* cent_preds = (const float*)d_in[2];    // (64,8732,1)
  const float* default_boxes = (const float*)d_in[3]; // (8732,4)
  const float* gt_boxes = (const float*)d_in[4];      // (64,16,4)
  const int* gt_labels = (const int*)d_in[5];         // (64,16)
  float* out = (float*)d_out;

  char* ws = (char*)d_ws;
  size_t off = 0;
  signed char* m = (signed char*)(ws + off);
  off += ((size_t)BNUM * PNUM + 255) & ~(size_t)255;
  int* bp = (int*)(ws + off);
  off += ((size_t)BNUM * MNUM * sizeof(int) + 255) & ~(size_t)255;
  float* rows = (float*)(ws + off);

  match_kernel<<<BNUM, 256, 0, stream>>>(default_boxes, gt_boxes, m, bp);
  override_kernel<<<BNUM, 1, 0, stream>>>(m, bp);
  loss_kernel<<<BNUM, 1024, 0, stream>>>(loc_preds, conf_preds, cent_preds,
                                         default_boxes, gt_boxes, gt_labels, m,
                                         rows);
  finalize_kernel<<<1, 64, 0, stream>>>(rows, out);
}